// L1Distance_72335839199292
// MI455X (gfx1250) — compile-verified
//
#include <hip/hip_runtime.h>
#include <hip/hip_bf16.h>

// L1 pairwise distance on MI455X (gfx1250).
// res[i,j] = sum_d |x1[i,d] - x2[j,d]|  (mean-adjustment cancels algebraically).
// VALU-bound (537M VALU ops vs ~17MB traffic) -> optimize for LDS reuse,
// async global->LDS staging (ASYNCcnt path), conflict-free ds_load_b128.

#define AS_GLOBAL __attribute__((address_space(1)))
#define AS_LDS    __attribute__((address_space(3)))

typedef int v4i __attribute__((ext_vector_type(4)));

#define TILE 64
#define PAD  68   // 68 floats = 272B row stride: 16B-aligned, bank offset 4/row
                  // -> 16 lanes reading rows (tx + 16*i) hit 64 distinct banks.

__global__ __launch_bounds__(256)
void l1dist_tile_kernel(const float* __restrict__ x1,
                        const float* __restrict__ x2,
                        float* __restrict__ out,
                        int N1, int N2) {
    __shared__ float x1s[TILE][TILE]; // stride 64: reads broadcast across lanes
    __shared__ float x2s[TILE][PAD];  // padded: conflict-free b128 lane reads

    const int t    = threadIdx.x;   // 0..255
    const int tx   = t & 15;        // col group
    const int ty   = t >> 4;        // row group
    const int col0 = blockIdx.x * TILE;
    const int row0 = blockIdx.y * TILE;

    // ---------------- Stage tiles into LDS ----------------
#if defined(__gfx1250__) && __has_builtin(__builtin_amdgcn_global_load_async_to_lds_b128)
    {
        // x1 tile: contiguous 64*64 floats starting at x1 + row0*64.
        const float* g1 = x1 + (size_t)row0 * 64;
        float*       l1 = &x1s[0][0];
#pragma unroll
        for (int k = 0; k < 4; ++k) {
            const int idx4 = t + 256 * k;      // float4 index, 1024 total
            __builtin_amdgcn_global_load_async_to_lds_b128(
                (AS_GLOBAL v4i*)(g1 + (size_t)idx4 * 4),
                (AS_LDS    v4i*)(l1 + (size_t)idx4 * 4),
                0, 0);
        }
        // x2 tile: row r, floats 16*seg..16*seg+3 -> x2s[r][same cols]; 64 rows * 4 segs = 256 b128 loads.
        const int r   = t >> 2;
        const int seg = t & 3;
        __builtin_amdgcn_global_load_async_to_lds_b128(
            (AS_GLOBAL v4i*)(x2 + (size_t)(col0 + r) * 64 + seg * 16),
            (AS_LDS    v4i*)(&x2s[r][seg * 16]),
            0, 0);
    }
  #if __has_builtin(__builtin_amdgcn_s_wait_asynccnt)
    __builtin_amdgcn_s_wait_asynccnt(0);
  #else
    asm volatile("s_wait_asynccnt 0" ::: "memory");
  #endif
#else
    {
        // Synchronous fallback (also used for host-side semantic pass).
        const float4* g1 = (const float4*)(x1 + (size_t)row0 * 64);
        float4*       l1 = (float4*)&x1s[0][0];
#pragma unroll
        for (int k = 0; k < 4; ++k) {
            const int idx4 = t + 256 * k;
            l1[idx4] = g1[idx4];
        }
        const int r   = t >> 2;
        const int seg = t & 3;
        *(float4*)&x2s[r][seg * 16] =
            *(const float4*)(x2 + (size_t)(col0 + r) * 64 + seg * 16);
    }
#endif
    __syncthreads();

    // ---------------- Compute 4x4 micro-tile per thread ----------------
    float acc[4][4];
#pragma unroll
    for (int j = 0; j < 4; ++j)
#pragma unroll
        for (int i = 0; i < 4; ++i)
            acc[j][i] = 0.0f;

#pragma unroll
    for (int d = 0; d < 64; d += 4) {
        float4 a[4], b[4];
#pragma unroll
        for (int j = 0; j < 4; ++j)
            a[j] = *(const float4*)&x1s[ty + 16 * j][d]; // broadcast read
#pragma unroll
        for (int i = 0; i < 4; ++i)
            b[i] = *(const float4*)&x2s[tx + 16 * i][d]; // conflict-free b128
#pragma unroll
        for (int j = 0; j < 4; ++j) {
#pragma unroll
            for (int i = 0; i < 4; ++i) {
                // Each term -> v_sub_f32 + v_add_f32 with |src| modifier.
                acc[j][i] += __builtin_fabsf(a[j].x - b[i].x)
                           + __builtin_fabsf(a[j].y - b[i].y)
                           + __builtin_fabsf(a[j].z - b[i].z)
                           + __builtin_fabsf(a[j].w - b[i].w);
            }
        }
    }

    // ---------------- Store (coalesced across tx) ----------------
#pragma unroll
    for (int j = 0; j < 4; ++j) {
        const int row = row0 + ty + 16 * j;
#pragma unroll
        for (int i = 0; i < 4; ++i) {
            const int col = col0 + tx + 16 * i;
            if (row < N1 && col < N2)
                out[(size_t)row * N2 + col] = fmaxf(acc[j][i], 0.0f);
        }
    }
}

extern "C" void kernel_launch(void* const* d_in, const int* in_sizes, int n_in,
                              void* d_out, int out_size, void* d_ws, size_t ws_size,
                              hipStream_t stream) {
    const float* x1 = (const float*)d_in[0];
    const float* x2 = (const float*)d_in[1];
    float* out = (float*)d_out;

    const int D  = 64;                       // fixed by the reference
    const int N1 = in_sizes[0] / D;          // 2048
    const int N2 = in_sizes[1] / D;          // 2048

    dim3 block(256);
    dim3 grid((N2 + TILE - 1) / TILE, (N1 + TILE - 1) / TILE);
    hipLaunchKernelGGL(l1dist_tile_kernel, grid, block, 0, stream,
                       x1, x2, out, N1, N2);
}